// Net_51196010168473
// MI455X (gfx1250) — compile-verified
//
#include <hip/hip_runtime.h>
#include <math.h>

// ---------------------------------------------------------------------------
// SplineConv GNN for MI455X (gfx1250, wave32).
//   * Edge message passing fused (no x_k materialization) -> scratch ~84 MB,
//     hot data (hidden states, agg, weights) stays resident in the 192MB L2.
//   * Dense root-transform GEMMs use V_WMMA_F32_16X16X4_F32 (f32-exact).
// ---------------------------------------------------------------------------

#define KSZ 5          // spline kernel size per pseudo dim
#define NGRAPH 128

typedef float v2f __attribute__((ext_vector_type(2)));
typedef float v8f __attribute__((ext_vector_type(8)));

// ---------------------------------------------------------------------------
// Generic zero fill
// ---------------------------------------------------------------------------
__global__ void zero_f32(float* __restrict__ p, int n) {
    int t = blockIdx.x * blockDim.x + threadIdx.x;
    if (t < n) p[t] = 0.0f;
}

// ---------------------------------------------------------------------------
// Per-edge spline basis (degree-1 open B-spline, 2-D pseudo): 4 combos/edge.
// Shared by all 3 layers.
// ---------------------------------------------------------------------------
__global__ void basis_kernel(const float* __restrict__ pseudo,
                             float* __restrict__ ebasis,
                             int* __restrict__ eidx, int E) {
    int e = blockIdx.x * blockDim.x + threadIdx.x;
    if (e >= E) return;
    float v0 = fminf(fmaxf(pseudo[2 * e + 0], 0.0f), 1.0f) * (float)(KSZ - 1);
    float v1 = fminf(fmaxf(pseudo[2 * e + 1], 0.0f), 1.0f) * (float)(KSZ - 1);
    int k0 = (int)floorf(v0); k0 = k0 < 0 ? 0 : (k0 > KSZ - 2 ? KSZ - 2 : k0);
    int k1 = (int)floorf(v1); k1 = k1 < 0 ? 0 : (k1 > KSZ - 2 ? KSZ - 2 : k1);
    float f0 = v0 - (float)k0;
    float f1 = v1 - (float)k1;
#pragma unroll
    for (int b0 = 0; b0 < 2; ++b0) {
#pragma unroll
        for (int b1 = 0; b1 < 2; ++b1) {
            int c = b0 * 2 + b1;
            float w0 = b0 ? f0 : (1.0f - f0);
            float w1 = b1 ? f1 : (1.0f - f1);
            ebasis[e * 4 + c] = w0 * w1;
            eidx[e * 4 + c] = (k0 + b0) * KSZ + (k1 + b1);
        }
    }
}

// ---------------------------------------------------------------------------
// In-degree (same for every layer; aggr='mean')
// ---------------------------------------------------------------------------
__global__ void deg_kernel(const int* __restrict__ dst, float* __restrict__ deg, int E) {
    int e = blockIdx.x * blockDim.x + threadIdx.x;
    if (e < E) atomicAdd(&deg[dst[e]], 1.0f);
}

// ---------------------------------------------------------------------------
// WMMA f32 GEMM: C[M,N] = A[M,KDIM] * B[KDIM,N], row-major everywhere.
// One wave32 per 16x16 C tile; V_WMMA_F32_16X16X4_F32 over KDIM/4 steps.
// Requires: M % 16 == 0 (M = 50000 = 3125*16), N % 16 == 0, KDIM % 4 == 0.
// A-frag layout (32-bit A 16x4): lanes 0-15 hold K=0/1 in v0/v1 for row M=lane;
//   lanes 16-31 hold K=2/3 for row M=lane-16.
// B-frag: lane&15 = column; lane-half + vgpr index selects K (mirror of A).
// C/D: vgpr j -> row (j + 8*laneHalf), col = lane&15.
// ---------------------------------------------------------------------------
template <int KDIM>
__global__ void wmma_gemm_f32(const float* __restrict__ A,
                              const float* __restrict__ B,
                              float* __restrict__ C, int N) {
    const int lane = threadIdx.x;       // 0..31, one wave per block
    const int half = lane >> 4;         // 0 or 1
    const int l16  = lane & 15;
    const int mbase = blockIdx.x * 16;
    const int nbase = blockIdx.y * 16;

    v8f c = {};
#pragma unroll
    for (int k0 = 0; k0 < KDIM; k0 += 4) {
        const int ka = k0 + (half ? 2 : 0);
        v2f a, b;
        a.x = A[(size_t)(mbase + l16) * KDIM + ka];
        a.y = A[(size_t)(mbase + l16) * KDIM + ka + 1];
        b.x = B[(size_t)ka * N + nbase + l16];
        b.y = B[(size_t)(ka + 1) * N + nbase + l16];
        c = __builtin_amdgcn_wmma_f32_16x16x4_f32(
                /*neg_a=*/false, a, /*neg_b=*/false, b,
                /*c_mod=*/(short)0, c, /*reuse_a=*/false, /*reuse_b=*/false);
    }
#pragma unroll
    for (int j = 0; j < 8; ++j) {
        C[(size_t)(mbase + j + (half ? 8 : 0)) * N + nbase + l16] = c[j];
    }
}

// ---------------------------------------------------------------------------
// Fused per-edge spline message + scatter-add.
// One block per edge; blockDim = OUT; x[src] staged in LDS once per edge.
// Weights (<=400 KB) and agg/x arrays are L2-resident on MI455X.
// ---------------------------------------------------------------------------
template <int IN, int OUT>
__global__ void edge_msg(const float* __restrict__ xin,
                         const float* __restrict__ w,      // [25, IN, OUT]
                         const int* __restrict__ src,
                         const int* __restrict__ dst,
                         const float* __restrict__ ebasis, // [E,4]
                         const int* __restrict__ eidx,     // [E,4]
                         float* __restrict__ agg, int E) {
    __shared__ float xs[IN];
    const int e = blockIdx.x;
    if (e >= E) return;
    const int tid = threadIdx.x;          // output channel
    const int s = src[e];
    for (int i = tid; i < IN; i += OUT) xs[i] = xin[(size_t)s * IN + i];
    __syncthreads();

    float acc = 0.0f;
#pragma unroll
    for (int c = 0; c < 4; ++c) {
        const float basis = ebasis[e * 4 + c];
        const int kidx = eidx[e * 4 + c];
        const float* wk = w + (size_t)kidx * IN * OUT + tid;
        float dot = 0.0f;
#pragma unroll
        for (int i = 0; i < IN; ++i) dot += xs[i] * wk[(size_t)i * OUT];
        acc += basis * dot;
    }
    atomicAdd(&agg[(size_t)dst[e] * OUT + tid], acc);
}

// ---------------------------------------------------------------------------
// mean-normalize + root transform (precomputed via WMMA) + bias + ELU
// ---------------------------------------------------------------------------
template <int OUT>
__global__ void finalize_layer(const float* __restrict__ agg,
                               const float* __restrict__ rootout,
                               const float* __restrict__ deg,
                               const float* __restrict__ bias,
                               float* __restrict__ hout, int n_nodes) {
    int t = blockIdx.x * blockDim.x + threadIdx.x;
    if (t >= n_nodes * OUT) return;
    int n = t / OUT, o = t % OUT;
    float d = fmaxf(deg[n], 1.0f);
    float v = agg[t] / d + rootout[t] + bias[o];
    hout[t] = v > 0.0f ? v : (expf(v) - 1.0f);   // ELU(alpha=1)
}

// ---------------------------------------------------------------------------
// Graph pooling (scatter-mean over batch)
// ---------------------------------------------------------------------------
__global__ void pool_kernel(const float* __restrict__ h,
                            const int* __restrict__ batch,
                            float* __restrict__ pooled, int n_nodes) {
    int t = blockIdx.x * blockDim.x + threadIdx.x;
    if (t >= n_nodes * 64) return;
    int n = t >> 6, o = t & 63;
    atomicAdd(&pooled[batch[n] * 64 + o], h[t]);
}

__global__ void cnt_kernel(const int* __restrict__ batch,
                           float* __restrict__ cnt, int n_nodes) {
    int t = blockIdx.x * blockDim.x + threadIdx.x;
    if (t < n_nodes) atomicAdd(&cnt[batch[t]], 1.0f);
}

// ---------------------------------------------------------------------------
// FC (64->30) + log_softmax, one thread per graph (128 total — trivial cost)
// ---------------------------------------------------------------------------
__global__ void fc_logsoftmax(const float* __restrict__ pooled,
                              const float* __restrict__ cnt,
                              const float* __restrict__ fcw,  // [64,30]
                              const float* __restrict__ fcb,  // [30]
                              float* __restrict__ out) {
    int g = blockIdx.x * blockDim.x + threadIdx.x;
    if (g >= NGRAPH) return;
    float c = fmaxf(cnt[g], 1.0f);
    float logits[30];
    float mx = -1e30f;
    for (int j = 0; j < 30; ++j) {
        float s = fcb[j];
        for (int o = 0; o < 64; ++o)
            s += (pooled[g * 64 + o] / c) * fcw[o * 30 + j];
        logits[j] = s;
        mx = fmaxf(mx, s);
    }
    float se = 0.0f;
    for (int j = 0; j < 30; ++j) se += expf(logits[j] - mx);
    float lse = mx + logf(se);
    for (int j = 0; j < 30; ++j) out[g * 30 + j] = logits[j] - lse;
}

// ---------------------------------------------------------------------------
// Host launcher
// ---------------------------------------------------------------------------
extern "C" void kernel_launch(void* const* d_in, const int* in_sizes, int n_in,
                              void* d_out, int out_size, void* d_ws, size_t ws_size,
                              hipStream_t stream) {
    const float* x      = (const float*)d_in[0];
    const int*   eix    = (const int*)  d_in[1];   // [2, E]
    const float* pseudo = (const float*)d_in[2];
    const int*   batch  = (const int*)  d_in[3];
    const float* w1 = (const float*)d_in[4];
    const float* r1 = (const float*)d_in[5];
    const float* b1 = (const float*)d_in[6];
    const float* w2 = (const float*)d_in[7];
    const float* r2 = (const float*)d_in[8];
    const float* b2 = (const float*)d_in[9];
    const float* w3 = (const float*)d_in[10];
    const float* r3 = (const float*)d_in[11];
    const float* b3 = (const float*)d_in[12];
    const float* fcw = (const float*)d_in[13];
    const float* fcb = (const float*)d_in[14];

    const int E = in_sizes[1] / 2;   // 800000
    const int N = in_sizes[3];       // 50000 (multiple of 16: 3125 tiles)
    const int* src = eix;
    const int* dst = eix + E;

    // ---- workspace carve-up (~84 MB total) ----
    char* ws = (char*)d_ws;
    size_t off = 0;
    auto carve = [&](size_t bytes) -> void* {
        void* p = ws + off;
        off = (off + bytes + 255) & ~(size_t)255;
        return p;
    };
    float* ebasis  = (float*)carve((size_t)E * 4 * sizeof(float));
    int*   eidx    = (int*)  carve((size_t)E * 4 * sizeof(int));
    float* deg     = (float*)carve((size_t)N * sizeof(float));
    float* h1      = (float*)carve((size_t)N * 32 * sizeof(float));
    float* h2      = (float*)carve((size_t)N * 64 * sizeof(float));
    float* h3      = (float*)carve((size_t)N * 64 * sizeof(float));
    float* agg     = (float*)carve((size_t)N * 64 * sizeof(float));  // reused
    float* rootout = (float*)carve((size_t)N * 64 * sizeof(float));  // reused
    float* pooled  = (float*)carve((size_t)NGRAPH * 64 * sizeof(float));
    float* cnt     = (float*)carve((size_t)NGRAPH * sizeof(float));
    (void)ws_size; (void)n_in; (void)out_size;

    const int T = 256;
    auto blocks = [](long long n, int t) { return (unsigned)((n + t - 1) / t); };

    // ---- shared precompute ----
    zero_f32<<<blocks(N, T), T, 0, stream>>>(deg, N);
    basis_kernel<<<blocks(E, T), T, 0, stream>>>(pseudo, ebasis, eidx, E);
    deg_kernel<<<blocks(E, T), T, 0, stream>>>(dst, deg, E);

    // ---- layer 1: 8 -> 32 ----
    zero_f32<<<blocks((long long)N * 32, T), T, 0, stream>>>(agg, N * 32);
    wmma_gemm_f32<8><<<dim3(N / 16, 32 / 16), 32, 0, stream>>>(x, r1, rootout, 32);
    edge_msg<8, 32><<<E, 32, 0, stream>>>(x, w1, src, dst, ebasis, eidx, agg, E);
    finalize_layer<32><<<blocks((long long)N * 32, T), T, 0, stream>>>(
        agg, rootout, deg, b1, h1, N);

    // ---- layer 2: 32 -> 64 ----
    zero_f32<<<blocks((long long)N * 64, T), T, 0, stream>>>(agg, N * 64);
    wmma_gemm_f32<32><<<dim3(N / 16, 64 / 16), 32, 0, stream>>>(h1, r2, rootout, 64);
    edge_msg<32, 64><<<E, 64, 0, stream>>>(h1, w2, src, dst, ebasis, eidx, agg, E);
    finalize_layer<64><<<blocks((long long)N * 64, T), T, 0, stream>>>(
        agg, rootout, deg, b2, h2, N);

    // ---- layer 3: 64 -> 64 ----
    zero_f32<<<blocks((long long)N * 64, T), T, 0, stream>>>(agg, N * 64);
    wmma_gemm_f32<64><<<dim3(N / 16, 64 / 16), 32, 0, stream>>>(h2, r3, rootout, 64);
    edge_msg<64, 64><<<E, 64, 0, stream>>>(h2, w3, src, dst, ebasis, eidx, agg, E);
    finalize_layer<64><<<blocks((long long)N * 64, T), T, 0, stream>>>(
        agg, rootout, deg, b3, h3, N);

    // ---- pooling + classifier ----
    zero_f32<<<blocks(NGRAPH * 64, T), T, 0, stream>>>(pooled, NGRAPH * 64);
    zero_f32<<<blocks(NGRAPH, T), T, 0, stream>>>(cnt, NGRAPH);
    pool_kernel<<<blocks((long long)N * 64, T), T, 0, stream>>>(h3, batch, pooled, N);
    cnt_kernel<<<blocks(N, T), T, 0, stream>>>(batch, cnt, N);
    fc_logsoftmax<<<1, NGRAPH, 0, stream>>>(pooled, cnt, fcw, fcb, (float*)d_out);
}